// Emb2_67748814127513
// MI455X (gfx1250) — compile-verified
//
#include <hip/hip_runtime.h>
#include <hip/hip_bf16.h>
#include <stdint.h>

typedef __attribute__((ext_vector_type(2))) float v2f;
typedef __attribute__((ext_vector_type(8))) float v8f;

#define N_ROWS_TOT 589824      // 768*768
#define K_ONES 32
#define DOUT 128
#define N_EX 16384
#define WAVES_PB 2
#define THREADS_PB (WAVES_PB * 32)
#define STAGE_FLOATS (K_ONES * DOUT)   // 4096 floats = 16KB per wave

__global__ __launch_bounds__(THREADS_PB) void emb_gather_sum_kernel(
    const float* __restrict__ tiles,   // (12,8,8,12,8,8,128) f32
    const float* __restrict__ zeros,   // (128,) f32 (all zero pad row)
    const int*   __restrict__ x,       // (16384,32) i32
    const int*   __restrict__ vflip,   // scalar
    float*       __restrict__ out)     // (16384,128) f32
{
  __shared__ __align__(16) float sm[WAVES_PB * STAGE_FLOATS + WAVES_PB * DOUT];

  const int lane = threadIdx.x & 31;
  const int wv   = threadIdx.x >> 5;
  const int e    = blockIdx.x * WAVES_PB + wv;   // one wave == one example
  const int flip = vflip[0];

  // ---- per-lane index transform (lane k owns hot-index k of this example) ----
  int r = x[e * K_ONES + lane];
  uint32_t packed;   // bit0 set -> use zeros row; else byte offset into tiles (mult of 512)
  if (r >= N_ROWS_TOT) {
    packed = 1u;                                  // pad row -> zeros
  } else {
    int f  = r & 7;
    int e2 = (r >> 3) & 7;
    int t  = r >> 6;                              // ((a*8+b)*8+c)*12 + d
    int d  = t % 12;
    int t2 = t / 12;                              // (a*8+b)*8 + c
    int c  = t2 & 7;
    int b  = (t2 >> 3) & 7;
    int a  = t2 >> 6;                             // 0..11
    bool z = (a == 0 && b == 0 && c == 3) || (d == 0 && e2 == 0 && f == 3);
    if (flip) {                                   // roll+6 (== -6 mod 12), rank flips
      a  = (a >= 6) ? a - 6 : a + 6;
      d  = (d >= 6) ? d - 6 : d + 6;
      b  = 7 - b;
      e2 = 7 - e2;
    }
    int src = ((((a * 8 + b) * 8 + c) * 12 + d) * 8 + e2) * 8 + f;
    packed = z ? 1u : ((uint32_t)src << 9);       // src * 512 bytes, bit0 free
  }

  // ---- async-gather 32 rows (512B each) into this wave's LDS region ----
  // One instruction per row: 32 lanes x 16B = full row, ASYNCcnt-tracked.
  float* smw = &sm[wv * STAGE_FLOATS];
  const uint32_t ldsWaveBase = (uint32_t)(size_t)(void*)smw;
  #pragma unroll 4
  for (int k = 0; k < K_ONES; ++k) {
    uint32_t p = __shfl(packed, k, 32);
    const char* base = (p & 1u) ? (const char*)zeros
                                : ((const char*)tiles + (p & ~1u));
    uint64_t ga = (uint64_t)(uintptr_t)base + (uint32_t)(lane * 16);
    uint32_t la = ldsWaveBase + (uint32_t)(k * 512 + lane * 16);
    asm volatile("global_load_async_to_lds_b128 %0, %1, off"
                 :: "v"(la), "v"(ga) : "memory");
  }
  asm volatile("s_wait_asynccnt 0x0" ::: "memory");

  // ---- WMMA column-sum: D = ones(16x4) x B(4x16) + C, 8 K-chunks per tile ----
  // A all-ones => D[m,n] = sum_k B[k,n]; row->halfwave mapping is irrelevant,
  // only lane%16 -> column N must be correct.
  const int col  = lane & 15;
  const int half = lane >> 4;
  v2f A; A.x = 1.0f; A.y = 1.0f;
  float* smres = &sm[WAVES_PB * STAGE_FLOATS + wv * DOUT];

  for (int n = 0; n < 8; ++n) {                   // 8 tiles of 16 columns
    v8f acc = {0.f, 0.f, 0.f, 0.f, 0.f, 0.f, 0.f, 0.f};
    #pragma unroll
    for (int k2 = 0; k2 < 8; ++k2) {              // 8 chunks of 4 rows
      int row = 4 * k2 + 2 * half;
      v2f B;
      B.x = smw[row * DOUT + n * 16 + col];
      B.y = smw[(row + 1) * DOUT + n * 16 + col];
      acc = __builtin_amdgcn_wmma_f32_16x16x4_f32(
                /*neg_a=*/false, A, /*neg_b=*/false, B,
                /*c_mod=*/(short)0, acc, /*reuse_a=*/false, /*reuse_b=*/false);
    }
    if (lane < 16) smres[n * 16 + col] = acc[0];  // M=0 row: lanes 0-15, VGPR0
  }

  // one coalesced 16B store per lane (same-wave LDS producer, DScnt-ordered)
  float4 o = ((const float4*)smres)[lane];
  ((float4*)(out + (size_t)e * DOUT))[lane] = o;
}

extern "C" void kernel_launch(void* const* d_in, const int* in_sizes, int n_in,
                              void* d_out, int out_size, void* d_ws, size_t ws_size,
                              hipStream_t stream) {
  const float* tiles = (const float*)d_in[0];
  const float* zeros = (const float*)d_in[1];
  const int*   x     = (const int*)d_in[2];
  const int*   vflip = (const int*)d_in[3];
  float*       out   = (float*)d_out;

  dim3 grid(N_EX / WAVES_PB);
  dim3 block(THREADS_PB);
  hipLaunchKernelGGL(emb_gather_sum_kernel, grid, block, 0, stream,
                     tiles, zeros, x, vflip, out);
}